// PointConv_9723805958814
// MI455X (gfx1250) — compile-verified
//
#include <hip/hip_runtime.h>

// ---------------------------------------------------------------------------
// PointConv for MI455X (gfx1250, wave32, WMMA f32 16x16x4)
//   B=8, N=4096, D=3, CIN=64, COUT=64, HID=32, CMCO=16, K=32
// All GEMM stages (MLP layers, pc contraction, final projection) run on the
// matrix pipe; VALU does only KNN distances and swish activations
// (fast path: v_exp_f32 + v_rcp_f32, no IEEE division sequence).
// ---------------------------------------------------------------------------

constexpr int N = 4096;

typedef float v2f __attribute__((ext_vector_type(2)));
typedef float v8f __attribute__((ext_vector_type(8)));

__device__ __forceinline__ float swishf(float x) {
  // x * sigmoid(x) with hardware v_rcp_f32 (~1 ulp) instead of IEEE divide
  return x * __builtin_amdgcn_rcpf(1.0f + __expf(-x));
}

__device__ __forceinline__ v8f wmma4(v2f a, v2f b, v8f c) {
  // D(16x16,f32) = A(16x4,f32) * B(4x16,f32) + C
  return __builtin_amdgcn_wmma_f32_16x16x4_f32(
      /*neg_a=*/false, a, /*neg_b=*/false, b,
      /*c_mod=*/(short)0, c, /*reuse_a=*/false, /*reuse_b=*/false);
}

// ---------------------------------------------------------------------------
// Kernel 1: brute-force 32-NN. One query per thread, candidate tiles in LDS,
// 32-best list per thread in LDS (stride 33 words -> conflict-free wave32).
// ---------------------------------------------------------------------------
__global__ __launch_bounds__(128) void knn_kernel(
    const float* __restrict__ xyz, const unsigned char* __restrict__ mask,
    int* __restrict__ knn) {
  __shared__ float sx[128], sy[128], sz[128];
  __shared__ int   smk[128];
  __shared__ float hd[128 * 33];
  __shared__ int   hi[128 * 33];

  const int tid = threadIdx.x;
  const int b   = blockIdx.x >> 5;                 // 32 blocks per batch
  const int q   = ((blockIdx.x & 31) << 7) + tid;  // query index
  const float* xb = xyz + (size_t)b * N * 3;

  const float qx = xb[q * 3 + 0];
  const float qy = xb[q * 3 + 1];
  const float qz = xb[q * 3 + 2];

  const int hb = tid * 33;
#pragma unroll
  for (int u = 0; u < 32; ++u) { hd[hb + u] = 3.0e38f; hi[hb + u] = 0; }
  float worst = 3.0e38f;
  int   wpos  = 0;

  for (int t0 = 0; t0 < N; t0 += 128) {
    __syncthreads();
    const int m = t0 + tid;
    sx[tid]  = xb[m * 3 + 0];
    sy[tid]  = xb[m * 3 + 1];
    sz[tid]  = xb[m * 3 + 2];
    smk[tid] = mask[(size_t)b * N + m] ? 1 : 0;
    __syncthreads();

    for (int jj = 0; jj < 128; ++jj) {
      const float dx = qx - sx[jj];
      const float dy = qy - sy[jj];
      const float dz = qz - sz[jj];
      float d = fmaf(dx, dx, fmaf(dy, dy, dz * dz));
      d = smk[jj] ? d : 1.0e8f;  // masked sources get BIG (still selectable)
      if (d < worst) {
        hd[hb + wpos] = d;
        hi[hb + wpos] = t0 + jj;
        float w = -1.0f; int wp = 0;
#pragma unroll
        for (int u = 0; u < 32; ++u) {
          const float v = hd[hb + u];
          if (v > w) { w = v; wp = u; }
        }
        worst = w; wpos = wp;
      }
    }
  }
#pragma unroll
  for (int u = 0; u < 32; ++u)
    knn[((size_t)b * N + q) * 32 + u] = hi[hb + u];
}

// ---------------------------------------------------------------------------
// Kernel 2: fully WMMA point-conv. One wave (32 lanes) per 8-point tile.
// Per point p (32 neighbor rows):
//   L1: h1(32x32) = swish(deltas(32x4) @ W1)        4 WMMA
//   L2: h2(32x32) = swish(h1 @ W2)                 32 WMMA
//   L3: w (32x16) = nm * swish(h2 @ W3)            16 WMMA
//   pc: pc^T(16x64) = w^T(16x32) @ v(32x64)        32 WMMA
// Then: out(8x64) = PC(8x1024) @ Wl(1024x64) + bl  1024 WMMA
// ---------------------------------------------------------------------------
#define HS   33    // activation row stride (words) -> conflict-free A reads
#define WROW 17    // s_wp row stride
#define VROW 65    // s_v  row stride -> conflict-free writes
#define PCS  1028  // s_pc row stride -> conflict-free A reads

__global__ __launch_bounds__(32) void pointconv_kernel(
    const float* __restrict__ xyz, const float* __restrict__ vals,
    const unsigned char* __restrict__ mask, const int* __restrict__ knn,
    const float* __restrict__ W1, const float* __restrict__ b1,
    const float* __restrict__ W2, const float* __restrict__ b2,
    const float* __restrict__ W3, const float* __restrict__ b3,
    const float* __restrict__ Wl, const float* __restrict__ bl,
    float* __restrict__ out) {
  __shared__ int   s_idx[8 * 32];
  __shared__ float s_q[24];
  __shared__ int   s_qm[8];
  __shared__ float s_bl[64];
  __shared__ float s_d[32 * 4];       // deltas [k][xyz0]
  __shared__ float s_nmf[32];         // neighbor mask as float
  __shared__ float s_h1[32 * HS];     // layer-1 activations
  __shared__ float s_h2[32 * HS];     // layer-2 activations
  __shared__ float s_wp[32 * WROW];   // w for one point: [k=32][d=16]
  __shared__ float s_v[32 * VROW];    // gathered vals:   [k=32][c=64]
  __shared__ float s_pc[8 * PCS];     // PC rows: [p=8][c*16+d = 1024]

  const int lane = threadIdx.x;
  const int lh   = lane >> 4;   // 0: K pair {0,1}, 1: K pair {2,3}
  const int l16  = lane & 15;
  const int k0h  = lh * 2;
  const int b    = blockIdx.x >> 9;          // 512 tiles per batch
  const int n0   = (blockIdx.x & 511) << 3;  // 8 points per tile

  // ---- MLP weight B-fragments + biases: held in registers for the block ----
  v2f   bW1[2], bW2[2][8], bW3[8];
  float bias1[2], bias2[2];
#pragma unroll
  for (int nt = 0; nt < 2; ++nt) {
    const int col = nt * 16 + l16;
    bW1[nt].x = (lh == 0) ? W1[0 * 32 + col] : W1[2 * 32 + col];
    bW1[nt].y = (lh == 0) ? W1[1 * 32 + col] : 0.0f;  // K=3 zero pad
    bias1[nt] = b1[col];
    bias2[nt] = b2[col];
#pragma unroll
    for (int kt = 0; kt < 8; ++kt) {
      const int k0 = kt * 4 + k0h;
      bW2[nt][kt].x = W2[(k0 + 0) * 32 + col];
      bW2[nt][kt].y = W2[(k0 + 1) * 32 + col];
    }
  }
#pragma unroll
  for (int kt = 0; kt < 8; ++kt) {
    const int k0 = kt * 4 + k0h;
    bW3[kt].x = W3[(k0 + 0) * 16 + l16];
    bW3[kt].y = W3[(k0 + 1) * 16 + l16];
  }
  const float bias3 = b3[l16];

  // ---- cooperative staging ----
  for (int t = lane; t < 64;  t += 32) s_bl[t] = bl[t];
  for (int t = lane; t < 256; t += 32)
    s_idx[t] = knn[((size_t)b * N + n0) * 32 + t];
  if (lane < 24) s_q[lane] = xyz[((size_t)b * N + n0) * 3 + lane];
  if (lane < 8)  s_qm[lane] = mask[(size_t)b * N + n0 + lane] ? 1 : 0;
  __builtin_prefetch(Wl + (size_t)lane * 2048, 0, 1);  // warm L2 for phase 3
  __syncthreads();

  const v8f z = {};

  for (int p = 0; p < 8; ++p) {
    // ---- per-lane staging: delta, neighbor mask, gathered vals row ----
    const int   j  = s_idx[p * 32 + lane];
    const float nm = mask[(size_t)b * N + j] ? 1.0f : 0.0f;
    const float* nb = xyz + ((size_t)b * N + j) * 3;
    s_d[lane * 4 + 0] = s_q[p * 3 + 0] - nb[0];
    s_d[lane * 4 + 1] = s_q[p * 3 + 1] - nb[1];
    s_d[lane * 4 + 2] = s_q[p * 3 + 2] - nb[2];
    s_d[lane * 4 + 3] = 0.0f;
    s_nmf[lane] = nm;

    const float4* vr = (const float4*)(vals + ((size_t)b * N + j) * 64);
#pragma unroll
    for (int c4 = 0; c4 < 16; ++c4) {
      float4 v4 = vr[c4];
      s_v[lane * VROW + c4 * 4 + 0] = v4.x * nm;
      s_v[lane * VROW + c4 * 4 + 1] = v4.y * nm;
      s_v[lane * VROW + c4 * 4 + 2] = v4.z * nm;
      s_v[lane * VROW + c4 * 4 + 3] = v4.w * nm;
    }
    __syncthreads();

    // ---- layer 1: h1 = swish(deltas @ W1 + b1), 2 M-tiles x 2 N-tiles ----
    v8f d1[2][2];
#pragma unroll
    for (int mt = 0; mt < 2; ++mt) {
      v2f a;
      a.x = s_d[(mt * 16 + l16) * 4 + k0h + 0];
      a.y = s_d[(mt * 16 + l16) * 4 + k0h + 1];
#pragma unroll
      for (int nt = 0; nt < 2; ++nt) d1[mt][nt] = wmma4(a, bW1[nt], z);
    }
#pragma unroll
    for (int mt = 0; mt < 2; ++mt)
#pragma unroll
      for (int nt = 0; nt < 2; ++nt)
#pragma unroll
        for (int r = 0; r < 8; ++r) {
          const int row = mt * 16 + r + lh * 8;
          s_h1[row * HS + nt * 16 + l16] = swishf(d1[mt][nt][r] + bias1[nt]);
        }
    __syncthreads();

    // ---- layer 2: h2 = swish(h1 @ W2 + b2) ----
    v8f d2[2][2];
#pragma unroll
    for (int mt = 0; mt < 2; ++mt)
#pragma unroll
      for (int nt = 0; nt < 2; ++nt) d2[mt][nt] = z;
#pragma unroll
    for (int mt = 0; mt < 2; ++mt)
#pragma unroll
      for (int kt = 0; kt < 8; ++kt) {
        v2f a;
        a.x = s_h1[(mt * 16 + l16) * HS + kt * 4 + k0h + 0];
        a.y = s_h1[(mt * 16 + l16) * HS + kt * 4 + k0h + 1];
#pragma unroll
        for (int nt = 0; nt < 2; ++nt)
          d2[mt][nt] = wmma4(a, bW2[nt][kt], d2[mt][nt]);
      }
#pragma unroll
    for (int mt = 0; mt < 2; ++mt)
#pragma unroll
      for (int nt = 0; nt < 2; ++nt)
#pragma unroll
        for (int r = 0; r < 8; ++r) {
          const int row = mt * 16 + r + lh * 8;
          s_h2[row * HS + nt * 16 + l16] = swishf(d2[mt][nt][r] + bias2[nt]);
        }
    __syncthreads();

    // ---- layer 3: w = nm * swish(h2 @ W3 + b3), stored k-major for pc ----
    v8f d3[2];
#pragma unroll
    for (int mt = 0; mt < 2; ++mt) d3[mt] = z;
#pragma unroll
    for (int mt = 0; mt < 2; ++mt)
#pragma unroll
      for (int kt = 0; kt < 8; ++kt) {
        v2f a;
        a.x = s_h2[(mt * 16 + l16) * HS + kt * 4 + k0h + 0];
        a.y = s_h2[(mt * 16 + l16) * HS + kt * 4 + k0h + 1];
        d3[mt] = wmma4(a, bW3[kt], d3[mt]);
      }
#pragma unroll
    for (int mt = 0; mt < 2; ++mt)
#pragma unroll
      for (int r = 0; r < 8; ++r) {
        const int krow = mt * 16 + r + lh * 8;  // neighbor index
        s_wp[krow * WROW + l16] = s_nmf[krow] * swishf(d3[mt][r] + bias3);
      }
    __syncthreads();

    // ---- pc^T = w^T(16x32) @ v(32x64) ----
#pragma unroll
    for (int nt = 0; nt < 4; ++nt) {
      v8f acc = z;
#pragma unroll
      for (int kt = 0; kt < 8; ++kt) {
        const int k0 = kt * 4 + k0h;
        v2f a, bb;
        a.x  = s_wp[(k0 + 0) * WROW + l16];           // A[M=l16][K=k0..]
        a.y  = s_wp[(k0 + 1) * WROW + l16];
        bb.x = s_v[(k0 + 0) * VROW + nt * 16 + l16];  // B[K][N=nt*16+l16]
        bb.y = s_v[(k0 + 1) * VROW + nt * 16 + l16];
        acc = wmma4(a, bb, acc);
      }
      // D element (M=d, N=c) -> PC[p][c*16+d]
#pragma unroll
      for (int r = 0; r < 8; ++r) {
        const int d = r + lh * 8;
        s_pc[p * PCS + (nt * 16 + l16) * 16 + d] = acc[r];
      }
    }
    __syncthreads();
  }

  // ---- out(8x64) = PC(8x1024) @ Wl(1024x64) + bl (rows 8..15 padded 0) ----
  v8f acc[4] = {};
  for (int kt = 0; kt < 256; ++kt) {
    const int k0 = kt * 4 + k0h;
    v2f a;
    a.x = (l16 < 8) ? s_pc[l16 * PCS + k0 + 0] : 0.0f;
    a.y = (l16 < 8) ? s_pc[l16 * PCS + k0 + 1] : 0.0f;
#pragma unroll
    for (int nt = 0; nt < 4; ++nt) {
      v2f bb;
      bb.x = Wl[(size_t)(k0 + 0) * 64 + nt * 16 + l16];
      bb.y = Wl[(size_t)(k0 + 1) * 64 + nt * 16 + l16];
      acc[nt] = wmma4(a, bb, acc[nt]);
    }
  }

  if (lh == 0) {  // lanes 0..15 hold rows M=0..7 in VGPRs 0..7
#pragma unroll
    for (int nt = 0; nt < 4; ++nt) {
#pragma unroll
      for (int r = 0; r < 8; ++r) {
        const int col = nt * 16 + l16;
        const float o = acc[nt][r] + s_bl[col];
        out[((size_t)b * N + n0 + r) * 64 + col] = s_qm[r] ? o : 0.0f;
      }
    }
  }
}

// ---------------------------------------------------------------------------
extern "C" void kernel_launch(void* const* d_in, const int* in_sizes, int n_in,
                              void* d_out, int out_size, void* d_ws,
                              size_t ws_size, hipStream_t stream) {
  (void)in_sizes; (void)n_in; (void)out_size; (void)ws_size;
  const float*         xyz  = (const float*)d_in[0];
  const float*         vals = (const float*)d_in[1];
  const unsigned char* mask = (const unsigned char*)d_in[2];
  const float* W1 = (const float*)d_in[3];
  const float* b1 = (const float*)d_in[4];
  const float* W2 = (const float*)d_in[5];
  const float* b2 = (const float*)d_in[6];
  const float* W3 = (const float*)d_in[7];
  const float* b3 = (const float*)d_in[8];
  const float* Wl = (const float*)d_in[9];
  const float* bl = (const float*)d_in[10];
  float* out = (float*)d_out;
  int*   knn = (int*)d_ws;  // [B=8][N=4096][K=32] int32 = 4 MB

  knn_kernel<<<256, 128, 0, stream>>>(xyz, mask, knn);
  pointconv_kernel<<<4096, 32, 0, stream>>>(xyz, vals, mask, knn,
                                            W1, b1, W2, b2, W3, b3, Wl, bl,
                                            out);
}